// GraphClassifier_11441792877013
// MI455X (gfx1250) — compile-verified
//
#include <hip/hip_runtime.h>
#include <math.h>

// ---------------------------------------------------------------------------
// GraphClassifier forward for MI455X (gfx1250, wave32).
//   K-selection : 4-pass radix select on |float| bits, exact index-order ties
//   GEMM1/GEMM2 : V_WMMA_F32_16X16X4_F32, two 16x16 C tiles per wave
//                 (independent accumulation chains for ILP), 32-bit offset
//                 addressing, global_prefetch_b8 on the weight stream,
//                 fused bias + ReLU + BatchNorm(eval) epilogue
//   Head        : tiny [100,1024]x[1024,2] dot + log_softmax, block reduce
// Memory-bound on W1 (82 MB read once, ~5us at 23.3 TB/s); f32 WMMA keeps
// the reference's f32 numerics over the K=40000 reduction.
// ---------------------------------------------------------------------------

#define B_SAMPLES 100
#define M_PAD     112      // 7 tiles of 16 rows
#define FLAT      40000
#define TOPK      12000
#define H1        512
#define H2        1024
#define BN_EPS    1e-5f

typedef __attribute__((ext_vector_type(2))) float v2f;
typedef __attribute__((ext_vector_type(8))) float v8f;

// ------------------------- Kernel 1: top-k mask ----------------------------
// One 256-thread block per (padded) sample row. Radix-select the K-th largest
// |adj| bit-pattern (monotone for non-negative floats), then emit
// h = keep ? x : 0 with exact lowest-index-first tie breaking and the
// reference's (vals != 0) exclusion.
__global__ __launch_bounds__(256) void topk_mask_kernel(
    const float* __restrict__ x, const float* __restrict__ adj,
    float* __restrict__ h)
{
  const int b   = blockIdx.x;
  const int tid = threadIdx.x;

  if (b >= B_SAMPLES) {               // zero the M-padding rows for the GEMMs
    float* row = h + (size_t)b * FLAT;
    for (int i = tid; i < FLAT; i += 256) row[i] = 0.0f;
    return;
  }

  const float* arow = adj + (size_t)b * FLAT;
  const float* xrow = x   + (size_t)b * FLAT;
  float*       hrow = h   + (size_t)b * FLAT;

  __shared__ unsigned s_hist[256];
  __shared__ unsigned s_scan[256];
  __shared__ unsigned s_prefix;
  __shared__ unsigned s_gt;

  if (tid == 0) { s_prefix = 0u; s_gt = 0u; }
  __syncthreads();

  // 4 radix passes, 8 bits each, high -> low (bit 31 is cleared by abs).
  for (int p = 0; p < 4; ++p) {
    const int shift = 24 - 8 * p;
    s_hist[tid] = 0u;
    __syncthreads();
    const unsigned prefix = s_prefix;
    for (int i = tid; i < FLAT; i += 256) {
      unsigned u = __float_as_uint(arow[i]) & 0x7fffffffu;
      bool in = (p == 0) || ((u >> (shift + 8)) == prefix);
      if (in) atomicAdd(&s_hist[(u >> shift) & 0xffu], 1u);
    }
    __syncthreads();
    if (tid == 0) {
      unsigned gt   = s_gt;
      unsigned need = (unsigned)TOPK - gt;   // >= 1 by construction
      unsigned cum  = 0;
      int j = 255;
      for (; j > 0; --j) {
        if (cum + s_hist[j] >= need) break;
        cum += s_hist[j];
      }
      s_gt     = gt + cum;                   // strictly-greater count so far
      s_prefix = (prefix << 8) | (unsigned)j;
    }
    __syncthreads();
  }

  const unsigned t        = s_prefix;            // K-th largest |value| bits
  const unsigned tie_keep = (unsigned)TOPK - s_gt;

  // Index-ordered tie ranking: contiguous chunks + block exclusive scan.
  const int chunk = (FLAT + 255) / 256;          // 157
  const int beg   = tid * chunk;
  const int end   = (beg + chunk < FLAT) ? (beg + chunk) : FLAT;

  unsigned cnt = 0;
  for (int i = beg; i < end; ++i) {
    unsigned u = __float_as_uint(arow[i]) & 0x7fffffffu;
    if (u == t) ++cnt;
  }
  s_scan[tid] = cnt;
  __syncthreads();
  if (tid == 0) {
    unsigned run = 0;
    for (int i = 0; i < 256; ++i) { unsigned c = s_scan[i]; s_scan[i] = run; run += c; }
  }
  __syncthreads();

  unsigned rank = s_scan[tid];
  for (int i = beg; i < end; ++i) {
    unsigned u = __float_as_uint(arow[i]) & 0x7fffffffu;
    bool keep;
    if (u > t) {
      keep = true;                               // u > t >= 0  =>  u != 0
    } else if (u == t) {
      keep = (rank < tie_keep) && (u != 0u);     // reference: vals!=0 only
      ++rank;
    } else {
      keep = false;
    }
    hrow[i] = keep ? xrow[i] : 0.0f;
  }
}

// ------------------ Kernel 2/3: WMMA GEMM + bias/ReLU/BN -------------------
// One wave (32 threads) per pair of 16x16 C tiles (N and N+16): shared A
// fragment, two independent WMMA accumulation chains. 32-bit element offsets
// against uniform bases (all arrays < 82 MB) to get SADDR+voffset addressing.
// ISA 7.12.2 layouts (wave32):
//   A frag: lane L -> (M = L%16, K = 2*(L/16)+{0,1})   -> one b64 load
//   B frag: lane L -> (N = L%16, K = 2*(L/16)+{0,1})   -> strided b32 loads
//   D vgpr r, lane L -> (m = r + 8*(L/16), n = L%16)
__global__ __launch_bounds__(32) void gemm_bn_relu_kernel(
    const float* __restrict__ A,     // [M_PAD, Kdim] row-major
    const float* __restrict__ Bw,    // [Kdim, Ndim]  row-major
    const float* __restrict__ bias,
    const float* __restrict__ gamma,
    const float* __restrict__ beta,
    const float* __restrict__ mean,
    const float* __restrict__ var,
    float* __restrict__ out,         // [M_PAD, Ndim]
    int Kdim, int Ndim)
{
  const int lane = threadIdx.x;      // wave32
  const int half = lane >> 4;        // 0 or 1
  const int lm   = lane & 15;
  const int m0   = blockIdx.x * 16;
  const int n0   = blockIdx.y * 32;  // two 16-wide N tiles per wave

  const unsigned uN = (unsigned)Ndim;
  unsigned aoff = (unsigned)(m0 + lm) * (unsigned)Kdim + 2u * (unsigned)half;
  unsigned boff = 2u * (unsigned)half * uN + (unsigned)(n0 + lm);
  const unsigned kstride = 32u * uN;

  v8f acc0 = {};
  v8f acc1 = {};
  for (int k = 0; k < Kdim; k += 32) {           // Kdim % 32 == 0 both layers
    // Stream weights ahead of use: lowers to global_prefetch_b8 (speculative,
    // may run past the end of Bw).
    __builtin_prefetch(Bw + (boff + 8u * kstride), 0, 1);
#pragma unroll
    for (int kk = 0; kk < 32; kk += 4) {
      v2f a = *(const v2f*)(A + (aoff + (unsigned)kk)); // A[m, k+2h..k+2h+1]
      const unsigned r0 = boff + (unsigned)kk * uN;     // row k+2h
      const unsigned r1 = r0 + uN;                      // row k+2h+1
      v2f b0, b1;
      b0.x = Bw[r0];        b0.y = Bw[r1];              // n tile 0
      b1.x = Bw[r0 + 16u];  b1.y = Bw[r1 + 16u];        // n tile 1
      acc0 = __builtin_amdgcn_wmma_f32_16x16x4_f32(
          false, a, false, b0, (short)0, acc0, false, false);
      acc1 = __builtin_amdgcn_wmma_f32_16x16x4_f32(
          false, a, false, b1, (short)0, acc1, false, false);
    }
    aoff += 32u;
    boff += kstride;
  }

  // Fused epilogue: y = (relu(z + bias) - mean) * g * rsqrt(v+eps) + beta
#pragma unroll
  for (int tile = 0; tile < 2; ++tile) {
    const int   n  = n0 + 16 * tile + lm;
    const float bi = bias[n];
    const float sc = gamma[n] * rsqrtf(var[n] + BN_EPS);
    const float sh = beta[n] - mean[n] * sc;
    const v8f   av = tile ? acc1 : acc0;
#pragma unroll
    for (int r = 0; r < 8; ++r) {
      const int m = m0 + r + 8 * half;
      float z = fmaxf(av[r] + bi, 0.0f);
      out[(unsigned)m * uN + (unsigned)n] = z * sc + sh;
    }
  }
}

// -------------------- Kernel 4: head + log_softmax -------------------------
__global__ __launch_bounds__(128) void head_kernel(
    const float* __restrict__ A2,    // [M_PAD, H2]
    const float* __restrict__ W3,    // [H2, 2]
    const float* __restrict__ b3,    // [2]
    float* __restrict__ out)         // [B_SAMPLES, 2]
{
  const int b   = blockIdx.x;        // sample (only 0..99 launched)
  const int tid = threadIdx.x;
  __shared__ float s0[128], s1[128];

  float a0 = 0.0f, a1 = 0.0f;
  const float* row = A2 + (size_t)b * H2;
  for (int j = tid; j < H2; j += 128) {
    float v = row[j];
    a0 = fmaf(v, W3[2 * j],     a0);
    a1 = fmaf(v, W3[2 * j + 1], a1);
  }
  s0[tid] = a0; s1[tid] = a1;
  __syncthreads();
  for (int s = 64; s > 0; s >>= 1) {
    if (tid < s) { s0[tid] += s0[tid + s]; s1[tid] += s1[tid + s]; }
    __syncthreads();
  }
  if (tid == 0) {
    float z0 = s0[0] + b3[0];
    float z1 = s1[0] + b3[1];
    float mx  = fmaxf(z0, z1);
    float lse = mx + logf(expf(z0 - mx) + expf(z1 - mx));
    out[2 * b]     = z0 - lse;
    out[2 * b + 1] = z1 - lse;
  }
}

// ------------------------------- launcher ----------------------------------
extern "C" void kernel_launch(void* const* d_in, const int* in_sizes, int n_in,
                              void* d_out, int out_size, void* d_ws, size_t ws_size,
                              hipStream_t stream) {
  const float* x   = (const float*)d_in[0];
  const float* adj = (const float*)d_in[1];
  const float* W1  = (const float*)d_in[2];
  const float* b1  = (const float*)d_in[3];
  const float* g1  = (const float*)d_in[4];
  const float* be1 = (const float*)d_in[5];
  const float* m1  = (const float*)d_in[6];
  const float* v1  = (const float*)d_in[7];
  const float* W2  = (const float*)d_in[8];
  const float* b2  = (const float*)d_in[9];
  const float* g2  = (const float*)d_in[10];
  const float* be2 = (const float*)d_in[11];
  const float* m2  = (const float*)d_in[12];
  const float* v2  = (const float*)d_in[13];
  const float* W3  = (const float*)d_in[14];
  const float* b3  = (const float*)d_in[15];
  float*       out = (float*)d_out;

  // Workspace layout (all re-initialized every call; ~18.6 MB total)
  char*  ws = (char*)d_ws;
  float* h  = (float*)ws;                                   // [M_PAD, FLAT]
  size_t off = (size_t)M_PAD * FLAT * sizeof(float);
  float* A1 = (float*)(ws + off);                           // [M_PAD, H1]
  off += (size_t)M_PAD * H1 * sizeof(float);
  float* A2 = (float*)(ws + off);                           // [M_PAD, H2]

  (void)in_sizes; (void)n_in; (void)out_size; (void)ws_size;

  topk_mask_kernel<<<M_PAD, 256, 0, stream>>>(x, adj, h);

  gemm_bn_relu_kernel<<<dim3(M_PAD / 16, H1 / 32), 32, 0, stream>>>(
      h, W1, b1, g1, be1, m1, v1, A1, FLAT, H1);

  gemm_bn_relu_kernel<<<dim3(M_PAD / 16, H2 / 32), 32, 0, stream>>>(
      A1, W2, b2, g2, be2, m2, v2, A2, H1, H2);

  head_kernel<<<B_SAMPLES, 128, 0, stream>>>(A2, W3, b3, out);
}